// TrackLoss_40166534152765
// MI455X (gfx1250) — compile-verified
//
#include <hip/hip_runtime.h>
#include <math.h>

// ---------------------------------------------------------------------------
// TrackLoss on MI455X (gfx1250, wave32)
//
// Score matrix: dist2_eff[k][o] = |key_k|^2 - 2*key_k.origin_o  (drop |o|^2:
// constant per origin row, argmin-invariant). Computed as a K=4 WMMA by
// augmenting the 2-D vectors:
//     A[k] = (-2kx, -2ky, |k|^2, 0)   (16x4, keys)
//     B[o] = ( ox,   oy,   1,   0)   (4x16, origins)
// One V_WMMA_F32_16X16X4_F32 evaluates 256 candidates.
//
// Per ISA 7.12.2 f32 layouts (wave32):
//   A: lane l -> row m=l%16; VGPR0/1 = K(0,1) for lanes 0-15, K(2,3) for 16-31
//   B: lane l -> col n=l%16; VGPR0/1 = K(0,1) for lanes 0-15, K(2,3) for 16-31
//   D: VGPR r, lane l -> row m = r + 8*(l>=16), col n = l%16
// => lane-local 8-way argmin per tile; one shfl_xor(16) merges the halves.
// ---------------------------------------------------------------------------

typedef __attribute__((ext_vector_type(2))) float v2f;
typedef __attribute__((ext_vector_type(8))) float v8f;

#define NPTS        32768
#define MKEYS       8192
#define TILES       (MKEYS / 16)          // 512 key tiles
#define CHUNK_TILES 128                   // 2048 keys per LDS chunk
#define NCHUNKS     (TILES / CHUNK_TILES) // 4
#define BLOCK       256                   // 8 wave32s
#define PTS_PER_BLOCK 128                 // 8 waves * 16 origin points

__global__ __launch_bounds__(BLOCK) void track_loss_init(float* ws) {
    ws[0] = 0.0f;   // weighted sum
    ws[1] = 0.0f;   // mask count
}

__global__ __launch_bounds__(BLOCK) void track_loss_main(
    const float* __restrict__ ocur,          // [N,2] flat_origin_curves
    const float* __restrict__ ncur,          // [N,2] flat_new_curves
    const float* __restrict__ dpts,          // [M,2] dict_points
    const unsigned char* __restrict__ dbool, // [M]   dict_bool (bool8)
    const float* __restrict__ dkeys,         // [M,2] dict_keys
    float* __restrict__ ws)
{
    // WMMA-ready A operands for a chunk of key tiles: 128 tiles * 32 lanes * 8B = 32 KB
    __shared__ v2f lds_a[CHUNK_TILES * 32];

    const int tid  = threadIdx.x;
    const int lane = tid & 31;
    const int wave = tid >> 5;
    const int col  = lane & 15;   // origin column within the 16x16 tile
    const int hi   = lane >> 4;   // 0: key rows 0-7 / K(0,1); 1: key rows 8-15 / K(2,3)

    // This wave's 16 origin points; lane l and l+16 share origin col.
    const int   n  = blockIdx.x * PTS_PER_BLOCK + wave * 16 + col;
    const float ox = ocur[2 * n + 0];
    const float oy = ocur[2 * n + 1];

    v2f b;                         // B operand: (ox,oy) low half, (1,0) high half
    b.x = hi ? 1.0f : ox;
    b.y = hi ? 0.0f : oy;

    float minval = 3.402823466e+38f;
    int   minidx = 0;
    const v8f czero = {};

    for (int c = 0; c < NCHUNKS; ++c) {
        // ---- Cooperative LDS fill: transform keys into per-lane A layout ----
        for (int e = tid; e < CHUNK_TILES * 32; e += BLOCK) {
            const int tc = e >> 5;         // tile within chunk
            const int s  = e & 31;         // lane slot
            const int k  = (c * CHUNK_TILES + tc) * 16 + (s & 15);
            const float kx = dkeys[2 * k + 0];
            const float ky = dkeys[2 * k + 1];
            v2f a;
            if (s < 16) { a.x = -2.0f * kx;       a.y = -2.0f * ky; }
            else        { a.x = kx * kx + ky * ky; a.y = 0.0f;      }
            lds_a[e] = a;
            // Warm L2/L0 for the next chunk's keys (global_prefetch_b8).
            if (c + 1 < NCHUNKS)
                __builtin_prefetch(&dkeys[2 * (k + CHUNK_TILES * 16)], 0, 0);
        }
        __syncthreads();

        // ---- WMMA sweep over this chunk's tiles + lane-local argmin ----
        for (int tc = 0; tc < CHUNK_TILES; ++tc) {
            const v2f a = lds_a[tc * 32 + lane];       // ds_load_b64
            const v8f d = __builtin_amdgcn_wmma_f32_16x16x4_f32(
                /*neg_a=*/false, a, /*neg_b=*/false, b,
                /*c_mod=*/(short)0, czero,
                /*reuse_a=*/false, /*reuse_b=*/false);
            const int kb = (c * CHUNK_TILES + tc) * 16 + hi * 8;
            #pragma unroll
            for (int r = 0; r < 8; ++r) {
                const float v  = d[r];
                const bool  lt = v < minval;           // strict < keeps first idx
                minval = lt ? v : minval;
                minidx = lt ? (kb + r) : minidx;
            }
        }
        __syncthreads();
    }

    // ---- Merge key-row halves (lane l <-> l^16 share an origin column) ----
    const float ov = __shfl_xor(minval, 16, 32);
    const int   oi = __shfl_xor(minidx, 16, 32);
    const bool take = (ov < minval) || (ov == minval && oi < minidx);
    minval = take ? ov : minval;
    minidx = take ? oi : minidx;

    // ---- Gather + masked per-point norm (lanes 0-15 own the 16 origins) ----
    float wsum = 0.0f, wcnt = 0.0f;
    if (hi == 0) {
        const int   idx = minidx;
        const float m   = dbool[idx] ? 1.0f : 0.0f;
        const float px  = dpts[2 * idx + 0];
        const float py  = dpts[2 * idx + 1];
        const float dx  = ncur[2 * n + 0] - px;
        const float dy  = ncur[2 * n + 1] - py;
        wsum = m * sqrtf(dx * dx + dy * dy);
        wcnt = m;
    }

    // ---- Wave reduction (lanes >=16 contribute zero), one atomic per wave ----
    #pragma unroll
    for (int off = 1; off < 16; off <<= 1) {
        wsum += __shfl_xor(wsum, off, 32);
        wcnt += __shfl_xor(wcnt, off, 32);
    }
    if (lane == 0) {
        atomicAdd(&ws[0], wsum);   // global_atomic_add_f32
        atomicAdd(&ws[1], wcnt);
    }
}

__global__ __launch_bounds__(BLOCK) void track_loss_final(const float* ws,
                                                          float* out) {
    out[0] = ws[0] / ws[1];
}

extern "C" void kernel_launch(void* const* d_in, const int* in_sizes, int n_in,
                              void* d_out, int out_size, void* d_ws, size_t ws_size,
                              hipStream_t stream) {
    const float*         ocur  = (const float*)d_in[0];
    const float*         ncur  = (const float*)d_in[1];
    const float*         dpts  = (const float*)d_in[2];
    const unsigned char* dbool = (const unsigned char*)d_in[3];
    const float*         dkeys = (const float*)d_in[4];
    float* out = (float*)d_out;
    float* ws  = (float*)d_ws;

    track_loss_init<<<1, 1, 0, stream>>>(ws);
    track_loss_main<<<NPTS / PTS_PER_BLOCK, BLOCK, 0, stream>>>(
        ocur, ncur, dpts, dbool, dkeys, ws);
    track_loss_final<<<1, 1, 0, stream>>>(ws, out);
}